// DUP_block_90391881712206
// MI455X (gfx1250) — compile-verified
//
#include <hip/hip_runtime.h>
#include <math.h>

#define NK 16
#define NBATCH 2048
#define NT 32
#define NRF 4
#define NG 16
#define BN_EPS 1e-3f
#define ALPHA 0.1f

typedef __attribute__((ext_vector_type(2))) float v2f;
typedef __attribute__((ext_vector_type(8))) float v8f;
typedef __attribute__((ext_vector_type(4))) unsigned int v4u;
typedef __attribute__((ext_vector_type(4))) int v4i;
typedef __attribute__((ext_vector_type(8))) int v8i;

#define AS1 __attribute__((address_space(1)))
#define AS3 __attribute__((address_space(3)))

// ---- CDNA5 data-movement feature detection (device pass only) ----
#if defined(__AMDGCN__) && __has_builtin(__builtin_amdgcn_tensor_load_to_lds)
#define HAVE_TDM 1
#else
#define HAVE_TDM 0
#endif

#if defined(__AMDGCN__) && \
    __has_builtin(__builtin_amdgcn_global_load_async_to_lds_b128)
#define HAVE_ASYNC 1
#else
#define HAVE_ASYNC 0
#endif

#if defined(__AMDGCN__) && __has_builtin(__builtin_amdgcn_s_wait_asynccnt)
#define WAIT_ASYNC_LE(n) __builtin_amdgcn_s_wait_asynccnt(n)
#else
#define WAIT_ASYNC_LE(n) asm volatile("s_wait_asynccnt " #n ::: "memory")
#endif

#if defined(__AMDGCN__) && __has_builtin(__builtin_amdgcn_s_wait_tensorcnt)
#define WAIT_TENSOR_LE(n) __builtin_amdgcn_s_wait_tensorcnt(n)
#else
#define WAIT_TENSOR_LE(n) asm volatile("s_wait_tensorcnt " #n ::: "memory")
#endif

#if HAVE_TDM
// 1-D TDM tile load: nfloats contiguous f32 from global -> LDS.
// D# group0: count=1, lds_addr, global_addr[56:0], type=2 ("image").
// D# group1: data_size=2 (4B), tensor_dim0 = tile_dim0 = nfloats,
//            tensor_dim1 = 1, tile_dim1 = 1, tensor_dim0_stride = nfloats.
static __device__ __forceinline__ void tdm_load_1d(const float* gsrc,
                                                   float* ldsdst, int nfloats) {
  const unsigned int lds_off =
      (unsigned int)(unsigned long long)(AS3 float*)ldsdst;
  const unsigned long long ga = (unsigned long long)(AS1 void*)(float*)gsrc;
  v4u g0;
  g0.x = 1u;                                   // count=1 (valid descriptor)
  g0.y = lds_off;                              // lds_addr (bytes)
  g0.z = (unsigned int)ga;                     // global_addr[31:0]
  g0.w = (unsigned int)((ga >> 32) & 0x01ffffffull) | 0x80000000u;  // +type=2
  const unsigned int nf = (unsigned int)nfloats;
  v8i g1;
  g1[0] = (int)(2u << 16);                     // wg_mask=0, data_size=4B
  g1[1] = (int)((nf & 0xffffu) << 16);         // tensor_dim0[15:0] @ bits63:48
  g1[2] = (int)(((nf >> 16) & 0xffffu) | (1u << 16));  // dim0 hi, tensor_dim1=1
  g1[3] = (int)((nf & 0xffffu) << 16);         // tile_dim0 @ bits127:112
  g1[4] = 1;                                   // tile_dim1=1, tile_dim2=0
  g1[5] = (int)nf;                             // tensor_dim0_stride[31:0]
  g1[6] = 0;
  g1[7] = 0;
  const v4i z4 = {0, 0, 0, 0};
#if __clang_major__ >= 23
  const v8i z8 = {0, 0, 0, 0, 0, 0, 0, 0};
  __builtin_amdgcn_tensor_load_to_lds(g0, g1, z4, z4, z8, 0);
#else
  __builtin_amdgcn_tensor_load_to_lds(g0, g1, z4, z4, 0);
#endif
}
#endif  // HAVE_TDM

static __device__ __forceinline__ v8f wmma4(v2f a, v2f b, v8f c) {
  // D = A(16x4) * B(4x16) + C(16x16), f32
  return __builtin_amdgcn_wmma_f32_16x16x4_f32(false, a, false, b, (short)0, c,
                                               false, false);
}

// complex accumulate: (cr + i*ci) += (ar + i*ai) @ (br + i*bi)
static __device__ __forceinline__ void cwmma(v8f& cr, v8f& ci, v2f ar, v2f ai,
                                             v2f br, v2f bi) {
  v2f nai = -ai;
  cr = wmma4(ar, br, cr);
  cr = wmma4(nai, bi, cr);
  ci = wmma4(ar, bi, ci);
  ci = wmma4(ai, br, ci);
}

// A fragment (16x4 f32): lane l -> m=l&15, hi=l>>4; elem v holds (r0+m, c0+2*hi+v)
// address = (r0+m)*rs + (c0+k)*cs
static __device__ __forceinline__ v2f frag_a(const float* S, int rs, int cs,
                                             int r0, int c0) {
  const int l = threadIdx.x;
  const int m = l & 15, hi = l >> 4;
  v2f a;
  a.x = S[(r0 + m) * rs + (c0 + 2 * hi + 0) * cs];
  a.y = S[(r0 + m) * rs + (c0 + 2 * hi + 1) * cs];
  return a;
}

// A fragment, unit column stride: K pair is contiguous & 8B aligned -> b64 load
static __device__ __forceinline__ v2f frag_a_c1(const float* S, int rs, int r0,
                                                int c0) {
  const int l = threadIdx.x;
  const int m = l & 15, hi = l >> 4;
  const float2 t =
      *reinterpret_cast<const float2*>(&S[(r0 + m) * rs + c0 + 2 * hi]);
  v2f a;
  a.x = t.x;
  a.y = t.y;
  return a;
}

// B fragment (4x16 f32): lane l -> n=l&15, hi=l>>4; elem v holds (r0+2*hi+v, c0+n)
// columns n >= ncols are zero-padded
static __device__ __forceinline__ v2f frag_b(const float* S, int rs, int cs,
                                             int r0, int c0, int ncols) {
  const int l = threadIdx.x;
  const int n = l & 15, hi = l >> 4;
  const bool ok = (n < ncols);
  const int nn = ok ? n : 0;
  v2f b;
  b.x = S[(r0 + 2 * hi + 0) * rs + (c0 + nn) * cs];
  b.y = S[(r0 + 2 * hi + 1) * rs + (c0 + nn) * cs];
  if (!ok) { b.x = 0.f; b.y = 0.f; }
  return b;
}

// In-place Gauss-Jordan inverse of a 16x16 complex matrix in LDS (no pivoting;
// inputs are Hermitian positive definite). One wave32, 8 elements per lane.
// Force-inlined so the LDS pointer arguments constant-fold per call site and
// stay in address space 3 (ds_* instructions, not flat). The pivot loop is
// barrier-separated and serially dependent -> keep it rolled (code size).
static __device__ __forceinline__ void cinv16(float* Ar, float* Ai,
                                              float* prow_r, float* prow_i,
                                              float* pcol_r, float* pcol_i,
                                              float* drec) {
  const int l = threadIdx.x;
#pragma clang loop unroll(disable)
  for (int p = 0; p < NG; ++p) {
    if (l < 16) {
      prow_r[l] = Ar[p * NG + l];
      prow_i[l] = Ai[p * NG + l];
    } else {
      const int r = l - 16;
      pcol_r[r] = Ar[r * NG + p];
      pcol_i[r] = Ai[r * NG + p];
    }
    __syncthreads();
    if (l == 0) {
      const float pr = prow_r[p], pi = prow_i[p];
      const float inv = 1.f / (pr * pr + pi * pi);
      drec[0] = pr * inv;
      drec[1] = -pi * inv;
    }
    __syncthreads();
    const float dr = drec[0], di = drec[1];
#pragma unroll
    for (int q = 0; q < 8; ++q) {
      const int e = l + 32 * q;
      const int r = e >> 4, j = e & 15;
      float or_, oi_;
      if (r == p && j == p) {
        or_ = dr; oi_ = di;
      } else if (r == p) {
        const float xr = prow_r[j], xi = prow_i[j];
        or_ = xr * dr - xi * di;
        oi_ = xr * di + xi * dr;
      } else if (j == p) {
        const float fr = pcol_r[r], fi = pcol_i[r];
        or_ = -(fr * dr - fi * di);
        oi_ = -(fr * di + fi * dr);
      } else {
        const float xr = prow_r[j], xi = prow_i[j];
        const float sr = xr * dr - xi * di;  // scaled pivot-row element
        const float si = xr * di + xi * dr;
        const float fr = pcol_r[r], fi = pcol_i[r];
        or_ = Ar[e] - (fr * sr - fi * si);
        oi_ = Ai[e] - (fr * si + fi * sr);
      }
      Ar[e] = or_;
      Ai[e] = oi_;
    }
    __syncthreads();
  }
}

__global__ __launch_bounds__(32) void riemann_precoder_kernel(
    const float* __restrict__ Frf_re, const float* __restrict__ Frf_im,
    const float* __restrict__ G_re, const float* __restrict__ G_im,
    const float* __restrict__ Lam_re, const float* __restrict__ Lam_im,
    const float* __restrict__ Beta_re, const float* __restrict__ bn_gamma,
    const float* __restrict__ bn_beta, const float* __restrict__ bn_mean,
    const float* __restrict__ bn_var, const float* __restrict__ dense_w,
    const float* __restrict__ dense_b, const float* __restrict__ P_mask,
    float* __restrict__ out) {
  // Double-buffered staging tiles (filled by TDM / async-LDS for iter k+1
  // while iter k computes). Lam buffer is inverted in place -> Li.
  __shared__ __align__(16) float sGhB_re[2][NT * NG], sGhB_im[2][NT * NG];
  __shared__ __align__(16) float sLamB_re[2][NG * NG], sLamB_im[2][NG * NG];
  __shared__ __align__(16) float sA_re[NG * NG], sA_im[NG * NG];  // M / Mi
  __shared__ __align__(16) float sGF_re[NG * NRF], sGF_im[NG * NRF];
  __shared__ __align__(16) float sT1_re[NG * NRF], sT1_im[NG * NRF];
  __shared__ __align__(16) float sT2_re[NG * NRF], sT2_im[NG * NRF];
  __shared__ __align__(16) float sFrf_re[NT * NRF], sFrf_im[NT * NRF];
  __shared__ __align__(16) float sEg_re[NT * NRF], sEg_im[NT * NRF];
  __shared__ float prow_r[NG], prow_i[NG], pcol_r[NG], pcol_i[NG];
  __shared__ float drec[2];
  __shared__ float sRed[32];

  const int b = blockIdx.x;
  const int l = threadIdx.x;
  const int n = l & 15, hi = l >> 4;

  // issue Gh (2 TDM) + Lam (4 async b128) tile loads for subcarrier kk -> buf
  auto stage = [&](int kk, int buf) {
    const size_t gb = ((size_t)kk * NBATCH + b) * (NT * NG);
    const size_t lb = ((size_t)kk * NBATCH + b) * (NG * NG);
#if HAVE_TDM
    tdm_load_1d(G_re + gb, &sGhB_re[buf][0], NT * NG);
    tdm_load_1d(G_im + gb, &sGhB_im[buf][0], NT * NG);
#elif HAVE_ASYNC
    {
      const int off = l * 4;  // 4 floats (16B) per lane per op
#pragma unroll
      for (int q = 0; q < 4; ++q) {
        const int e = q * 128 + off;
        __builtin_amdgcn_global_load_async_to_lds_b128(
            (v4i*)(G_re + gb + e), (v4i*)(&sGhB_re[buf][0] + e), 0, 0);
        __builtin_amdgcn_global_load_async_to_lds_b128(
            (v4i*)(G_im + gb + e), (v4i*)(&sGhB_im[buf][0] + e), 0, 0);
      }
    }
#else
#pragma unroll
    for (int q = 0; q < 16; ++q) {
      const int e = l + 32 * q;
      sGhB_re[buf][e] = G_re[gb + e];
      sGhB_im[buf][e] = G_im[gb + e];
    }
#endif
#if HAVE_ASYNC
    {
      const int off = l * 4;
#pragma unroll
      for (int q = 0; q < 2; ++q) {
        const int e = q * 128 + off;
        __builtin_amdgcn_global_load_async_to_lds_b128(
            (v4i*)(Lam_re + lb + e), (v4i*)(&sLamB_re[buf][0] + e), 0, 0);
        __builtin_amdgcn_global_load_async_to_lds_b128(
            (v4i*)(Lam_im + lb + e), (v4i*)(&sLamB_im[buf][0] + e), 0, 0);
      }
    }
#elif HAVE_TDM
    tdm_load_1d(Lam_re + lb, &sLamB_re[buf][0], NG * NG);
    tdm_load_1d(Lam_im + lb, &sLamB_im[buf][0], NG * NG);
#else
#pragma unroll
    for (int q = 0; q < 8; ++q) {
      const int e = l + 32 * q;
      sLamB_re[buf][e] = Lam_re[lb + e];
      sLamB_im[buf][e] = Lam_im[lb + e];
    }
#endif
  };

  // ---- load Frf[b], zero eg accumulator; issue first tile fetch ----
  stage(0, 0);
#pragma unroll
  for (int q = 0; q < 4; ++q) {
    const int e = l + 32 * q;
    sFrf_re[e] = Frf_re[(size_t)b * NT * NRF + e];
    sFrf_im[e] = Frf_im[(size_t)b * NT * NRF + e];
    sEg_re[e] = 0.f;
    sEg_im[e] = 0.f;
  }
  __syncthreads();

  // ---- learned step size: row-sums -> BN -> Dense(1) -> leaky_relu ----
  float part = 0.f;
#pragma unroll
  for (int q = 0; q < 2; ++q) {
    const int f = l + 32 * q;  // feature index in [0,64)
    const int t = f & 31;
    const float* src = (f < NT) ? sFrf_re : sFrf_im;
    const float rsum = src[t * NRF + 0] + src[t * NRF + 1] +
                       src[t * NRF + 2] + src[t * NRF + 3];
    const float h = bn_gamma[f] * (rsum - bn_mean[f]) *
                        rsqrtf(bn_var[f] + BN_EPS) + bn_beta[f];
    part += h * dense_w[f];
  }
  sRed[l] = part;
  __syncthreads();
  if (l == 0) {
    float s = dense_b[0];
    for (int i = 0; i < 32; ++i) s += sRed[i];
    sRed[0] = (s > 0.f) ? s : ALPHA * s;
  }
  __syncthreads();
  const float step = sRed[0];
  __syncthreads();

  // ---- accumulate Euclidean gradient over Nk subcarriers (dbl-buffered) ----
  // Unroll by 2 so the buffer index c = k&1 is a compile-time constant in each
  // copy: LDS bases stay constant GEPs into __shared__ arrays -> ds_* (not
  // flat_*) instructions after address-space inference.
#pragma unroll 2
  for (int k = 0; k < NK; ++k) {
    const int c = k & 1;

    // issue next tile; wait for current one (in-order counters: allow only
    // the newest ops to remain outstanding)
    if (k + 1 < NK) {
      stage(k + 1, c ^ 1);
#if HAVE_TDM
      WAIT_TENSOR_LE(2);
#endif
#if HAVE_ASYNC
      WAIT_ASYNC_LE(4);
#endif
    } else {
#if HAVE_TDM
      WAIT_TENSOR_LE(0);
#endif
#if HAVE_ASYNC
      WAIT_ASYNC_LE(0);
#endif
    }
    if (k + 2 < NK) {  // warm L2 ahead of the TDM fetch after next
      const size_t gp = ((size_t)(k + 2) * NBATCH + b) * (NT * NG);
      __builtin_prefetch(&G_re[gp], 0, 1);
      __builtin_prefetch(&G_im[gp], 0, 1);
    }
    const float invBeta = 1.f / Beta_re[(size_t)k * NBATCH + b];
    __syncthreads();

    const float* Gr = &sGhB_re[c][0];
    const float* Gi = &sGhB_im[c][0];
    float* Lir = &sLamB_re[c][0];  // Lam -> (in place) -> Li
    float* Lii = &sLamB_im[c][0];

    // Li = inv(Lam), in place in the staging buffer
    cinv16(Lir, Lii, prow_r, prow_i, pcol_r, pcol_i, drec);

    // GF(16x4) = conj(Gh)^T(16x32) @ Frf(32x4)   [K=32 -> 8 chunks]
    {
      v8f cr{}, ci{};
#pragma unroll
      for (int kk = 0; kk < 8; ++kk) {
        v2f ar = frag_a(Gr, 1, NG, 0, kk * 4);   // G[g][t] addr = t*NG+g
        v2f ai = -frag_a(Gi, 1, NG, 0, kk * 4);  // conj
        v2f br = frag_b(sFrf_re, NRF, 1, kk * 4, 0, NRF);
        v2f bi = frag_b(sFrf_im, NRF, 1, kk * 4, 0, NRF);
        cwmma(cr, ci, ar, ai, br, bi);
      }
      if (n < NRF) {
#pragma unroll
        for (int v = 0; v < 8; ++v) {
          sGF_re[(v + 8 * hi) * NRF + n] = cr[v];
          sGF_im[(v + 8 * hi) * NRF + n] = ci[v];
        }
      }
    }
    __syncthreads();

    // M(16x16) = Li + GF @ GF^H * invBeta   [K=4, single chunk]
    {
      v8f mr{}, mi{};
      v2f ar = frag_a_c1(sGF_re, NRF, 0, 0);
      v2f ai = frag_a_c1(sGF_im, NRF, 0, 0);
      v2f br = frag_b(sGF_re, 1, NRF, 0, 0, 16);   // GF^H[k][n] = conj(GF[n][k])
      v2f bi = -frag_b(sGF_im, 1, NRF, 0, 0, 16);
      cwmma(mr, mi, ar, ai, br, bi);
#pragma unroll
      for (int v = 0; v < 8; ++v) {
        const int rr = v + 8 * hi;
        sA_re[rr * NG + n] = Lir[rr * NG + n] + mr[v] * invBeta;
        sA_im[rr * NG + n] = Lii[rr * NG + n] + mi[v] * invBeta;
      }
    }
    __syncthreads();

    // Mi = inv(M)  (in place in sA)
    cinv16(sA_re, sA_im, prow_r, prow_i, pcol_r, pcol_i, drec);

    // T1(16x4) = Mi @ GF   [K=16 -> 4 chunks]
    {
      v8f cr{}, ci{};
#pragma unroll
      for (int kk = 0; kk < 4; ++kk) {
        v2f ar = frag_a_c1(sA_re, NG, 0, kk * 4);
        v2f ai = frag_a_c1(sA_im, NG, 0, kk * 4);
        v2f br = frag_b(sGF_re, NRF, 1, kk * 4, 0, NRF);
        v2f bi = frag_b(sGF_im, NRF, 1, kk * 4, 0, NRF);
        cwmma(cr, ci, ar, ai, br, bi);
      }
      if (n < NRF) {
#pragma unroll
        for (int v = 0; v < 8; ++v) {
          sT1_re[(v + 8 * hi) * NRF + n] = cr[v];
          sT1_im[(v + 8 * hi) * NRF + n] = ci[v];
        }
      }
    }
    __syncthreads();

    // T2(16x4) = Mi @ T1
    {
      v8f cr{}, ci{};
#pragma unroll
      for (int kk = 0; kk < 4; ++kk) {
        v2f ar = frag_a_c1(sA_re, NG, 0, kk * 4);
        v2f ai = frag_a_c1(sA_im, NG, 0, kk * 4);
        v2f br = frag_b(sT1_re, NRF, 1, kk * 4, 0, NRF);
        v2f bi = frag_b(sT1_im, NRF, 1, kk * 4, 0, NRF);
        cwmma(cr, ci, ar, ai, br, bi);
      }
      if (n < NRF) {
#pragma unroll
        for (int v = 0; v < 8; ++v) {
          sT2_re[(v + 8 * hi) * NRF + n] = cr[v];
          sT2_im[(v + 8 * hi) * NRF + n] = ci[v];
        }
      }
    }
    __syncthreads();

    // eg_k(32x4) = Gh @ T2 * invBeta ; accumulate (two 16-row halves)
#pragma unroll
    for (int h2 = 0; h2 < 2; ++h2) {
      v8f er{}, ei{};
#pragma unroll
      for (int kk = 0; kk < 4; ++kk) {
        v2f ar = frag_a_c1(Gr, NG, h2 * 16, kk * 4);
        v2f ai = frag_a_c1(Gi, NG, h2 * 16, kk * 4);
        v2f br = frag_b(sT2_re, NRF, 1, kk * 4, 0, NRF);
        v2f bi = frag_b(sT2_im, NRF, 1, kk * 4, 0, NRF);
        cwmma(er, ei, ar, ai, br, bi);
      }
      if (n < NRF) {
#pragma unroll
        for (int v = 0; v < 8; ++v) {
          const int t = h2 * 16 + v + 8 * hi;
          sEg_re[t * NRF + n] += er[v] * invBeta;
          sEg_im[t * NRF + n] += ei[v] * invBeta;
        }
      }
    }
    __syncthreads();
  }

  // ---- eg = -mean_k * mask ; Riemannian grad ; norm ; update ; clamp ----
  float rg_r[4], rg_i[4], fr_[4], fi_[4];
  float nrm_part = 0.f;
#pragma unroll
  for (int q = 0; q < 4; ++q) {
    const int e = l + 32 * q;  // e = t*NRF + r
    const float mask = P_mask[e];
    const float er = -(sEg_re[e] * (1.f / NK)) * mask;
    const float ei = -(sEg_im[e] * (1.f / NK)) * mask;
    const float fr = sFrf_re[e], fi = sFrf_im[e];
    const float proj = er * fr + ei * fi;  // Re(eg * conj(Frf))
    const float rr = er - proj * fr;
    const float ri = ei - proj * fi;
    rg_r[q] = rr; rg_i[q] = ri; fr_[q] = fr; fi_[q] = fi;
    nrm_part += rr * rr + ri * ri;
  }
  sRed[l] = nrm_part;
  __syncthreads();
  if (l == 0) {
    float s = 0.f;
    for (int i = 0; i < 32; ++i) s += sRed[i];
    sRed[0] = sqrtf(s);
  }
  __syncthreads();
  const float sc = step / sRed[0];
#pragma unroll
  for (int q = 0; q < 4; ++q) {
    const int e = l + 32 * q;
    const float xr = fr_[q] - sc * rg_r[q];
    const float xi = fi_[q] - sc * rg_i[q];
    const float mag = sqrtf(xr * xr + xi * xi);
    float ex = mag - 1.f;
    if (ex < 0.f) ex = 0.f;
    const float scale = ex + 1.f;
    const size_t o = ((size_t)b * NT * NRF + e) * 2;  // complex64 interleaved
    out[o + 0] = xr / scale;
    out[o + 1] = xi / scale;
  }
}

extern "C" void kernel_launch(void* const* d_in, const int* in_sizes, int n_in,
                              void* d_out, int out_size, void* d_ws,
                              size_t ws_size, hipStream_t stream) {
  (void)in_sizes; (void)n_in; (void)out_size; (void)d_ws; (void)ws_size;
  const float* Frf_re   = (const float*)d_in[0];
  const float* Frf_im   = (const float*)d_in[1];
  const float* G_re     = (const float*)d_in[2];
  const float* G_im     = (const float*)d_in[3];
  const float* Lam_re   = (const float*)d_in[4];
  const float* Lam_im   = (const float*)d_in[5];
  const float* Beta_re  = (const float*)d_in[6];
  const float* bn_gamma = (const float*)d_in[7];
  const float* bn_beta  = (const float*)d_in[8];
  const float* bn_mean  = (const float*)d_in[9];
  const float* bn_var   = (const float*)d_in[10];
  const float* dense_w  = (const float*)d_in[11];
  const float* dense_b  = (const float*)d_in[12];
  const float* P_mask   = (const float*)d_in[13];
  float* out = (float*)d_out;

  riemann_precoder_kernel<<<dim3(NBATCH), dim3(32), 0, stream>>>(
      Frf_re, Frf_im, G_re, G_im, Lam_re, Lam_im, Beta_re, bn_gamma, bn_beta,
      bn_mean, bn_var, dense_w, dense_b, P_mask, out);
}